// GCNLayer_27608049778855
// MI455X (gfx1250) — compile-verified
//
#include <hip/hip_runtime.h>

#define IN_DIM 64
#define OUT_DIM 64

typedef float v2f __attribute__((ext_vector_type(2)));
typedef float v8f __attribute__((ext_vector_type(8)));

// ---------------------------------------------------------------------------
// Zeroing kernels (accumulators must be zero at the start of every launch).
// ---------------------------------------------------------------------------
__global__ void zero4_kernel(float4* __restrict__ p, int n4) {
    int i = blockIdx.x * blockDim.x + threadIdx.x;
    int stride = gridDim.x * blockDim.x;
    float4 z = make_float4(0.f, 0.f, 0.f, 0.f);
    for (; i < n4; i += stride) p[i] = z;
}

__global__ void zero1_kernel(float* __restrict__ p, int n) {
    int i = blockIdx.x * blockDim.x + threadIdx.x;
    int stride = gridDim.x * blockDim.x;
    for (; i < n; i += stride) p[i] = 0.0f;
}

// ---------------------------------------------------------------------------
// Edge scatter: msg[dst] += x[src] (per feature), deg[dst] += 1.
// One thread per (edge, feature). Consecutive lanes = consecutive features of
// the same edge -> fully coalesced gather + atomics on 1-2 cache lines.
// ---------------------------------------------------------------------------
__global__ void edge_scatter_kernel(const float* __restrict__ x,
                                    const int* __restrict__ src,
                                    const int* __restrict__ dst,
                                    float* __restrict__ msg,   // aliases d_out
                                    float* __restrict__ deg,
                                    int nEdges) {
    int gid = blockIdx.x * blockDim.x + threadIdx.x;
    int total = nEdges * IN_DIM;                 // 76.8M, fits in int32
    if (gid >= total) return;
    int e = gid >> 6;                            // edge index
    int f = gid & (IN_DIM - 1);                  // feature index
    int s = src[e];
    int d = dst[e];
    atomicAdd(&msg[(size_t)d * IN_DIM + f], x[(size_t)s * IN_DIM + f]);
    if (f == 0) atomicAdd(&deg[d], 1.0f);
}

// ---------------------------------------------------------------------------
// Mean + Linear via full-precision WMMA: out = (msg/deg) @ W^T + b.
// One wave per 16-row tile; 4 f32 16x16 accumulators cover all 64 out feats.
// msg aliases out: each wave loads its entire 16x64 A tile into registers
// BEFORE storing (data dependency through the WMMA chain guarantees order).
// ---------------------------------------------------------------------------
#define LDS_STRIDE (IN_DIM + 1)   // pad to avoid LDS bank conflicts

__global__ void gcn_gemm_kernel(const float* __restrict__ W,    // [OUT][IN]
                                const float* __restrict__ bias, // [OUT]
                                const float* __restrict__ deg,  // [N]
                                float* hout,                    // msg in, out out
                                int nRowTiles) {
    __shared__ float sW[OUT_DIM * LDS_STRIDE];

    // Stage W into LDS once per block (shared by 8 waves).
    for (int i = threadIdx.x; i < OUT_DIM * IN_DIM; i += blockDim.x) {
        int n = i >> 6;
        int k = i & (IN_DIM - 1);
        sW[n * LDS_STRIDE + k] = W[i];
    }
    __syncthreads();

    const int lane = threadIdx.x & 31;
    const int waveInBlock = threadIdx.x >> 5;
    const int wavesPerBlock = blockDim.x >> 5;
    const int rowTile = blockIdx.x * wavesPerBlock + waveInBlock;
    if (rowTile >= nRowTiles) return;            // wave-uniform exit

    const int row0 = rowTile * 16;
    const int m    = lane & 15;                  // matrix row (A) / col (B,C)
    const int hi   = lane >> 4;                  // lane-half selector
    const int koff = hi * 2;                     // K offset within group of 4

    // Per-lane row scale: 1 / max(deg, 1)
    const int r = row0 + m;
    float dv  = deg[r];
    float inv = 1.0f / fmaxf(dv, 1.0f);

    // Preload ALL A fragments (16 K-steps) before any store (out aliases msg).
    // A 16x4 f32 layout: VGPR0 = K {0 | 2}, VGPR1 = K {1 | 3} by lane half.
    v2f a[16];
    const float* rowp = hout + (size_t)r * IN_DIM;
#pragma unroll
    for (int kk = 0; kk < 16; ++kk) {
        a[kk].x = rowp[kk * 4 + koff]     * inv;
        a[kk].y = rowp[kk * 4 + koff + 1] * inv;
    }

    v8f acc[4] = {};                             // zero accumulators

    // B[k][n] = W^T[k][n] = W[n][k]; lane's column n = t*16 + m.
#pragma unroll
    for (int kk = 0; kk < 16; ++kk) {
#pragma unroll
        for (int t = 0; t < 4; ++t) {
            int n = t * 16 + m;
            v2f b;
            b.x = sW[n * LDS_STRIDE + kk * 4 + koff];
            b.y = sW[n * LDS_STRIDE + kk * 4 + koff + 1];
            acc[t] = __builtin_amdgcn_wmma_f32_16x16x4_f32(
                /*neg_a=*/false, a[kk], /*neg_b=*/false, b,
                /*c_mod=*/(short)0, acc[t],
                /*reuse_a=*/false, /*reuse_b=*/false);
        }
    }

    // Bias + store. C/D layout: VGPR v -> row (v | v+8) by lane half, col = m.
#pragma unroll
    for (int t = 0; t < 4; ++t) {
        int n = t * 16 + m;
        float bv = bias[n];
#pragma unroll
        for (int v = 0; v < 8; ++v) {
            int rr = row0 + v + hi * 8;
            hout[(size_t)rr * OUT_DIM + n] = acc[t][v] + bv;
        }
    }
}

// ---------------------------------------------------------------------------
// Launcher
// ---------------------------------------------------------------------------
extern "C" void kernel_launch(void* const* d_in, const int* in_sizes, int n_in,
                              void* d_out, int out_size, void* d_ws, size_t ws_size,
                              hipStream_t stream) {
    const float* x   = (const float*)d_in[0];
    const int*   src = (const int*)  d_in[1];
    const int*   dst = (const int*)  d_in[2];
    const float* W   = (const float*)d_in[3];
    const float* b   = (const float*)d_in[4];

    float* out = (float*)d_out;      // doubles as the msg-sum accumulator
    float* deg = (float*)d_ws;       // N_NODES floats of scratch

    const int nNodes = in_sizes[0] / IN_DIM;   // 100000
    const int nEdges = in_sizes[1];            // 1200000

    // 1) Zero the accumulators.
    {
        int n4 = (nNodes * OUT_DIM) / 4;
        int blocks = (n4 + 255) / 256;
        if (blocks > 4096) blocks = 4096;
        zero4_kernel<<<blocks, 256, 0, stream>>>((float4*)out, n4);
    }
    {
        int blocks = (nNodes + 255) / 256;
        if (blocks > 2048) blocks = 2048;
        zero1_kernel<<<blocks, 256, 0, stream>>>(deg, nNodes);
    }

    // 2) Edge scatter (atomic accumulate into out, degrees into ws).
    {
        int total = nEdges * IN_DIM;
        int blocks = (total + 255) / 256;
        edge_scatter_kernel<<<blocks, 256, 0, stream>>>(x, src, dst, out, deg, nEdges);
    }

    // 3) Mean + linear layer via fp32 WMMA (in-place on d_out).
    {
        int nRowTiles = (nNodes + 15) / 16;    // 6250
        int wavesPerBlock = 8;                 // 256 threads
        int blocks = (nRowTiles + wavesPerBlock - 1) / wavesPerBlock;
        gcn_gemm_kernel<<<blocks, 256, 0, stream>>>(W, b, deg, out, nRowTiles);
    }
}